// SpacyLoss_46213848105408
// MI455X (gfx1250) — compile-verified
//
#include <hip/hip_runtime.h>
#include <hip/hip_bf16.h>

typedef float v2f __attribute__((ext_vector_type(2)));
typedef float v8f __attribute__((ext_vector_type(8)));

#define N_TOT   65536
#define BC      64
#define K_REAL  110
#define KP      112            // K padded to 7 tiles of 16
#define WS_X    0              // ws[0 .. 7168) = X (64 x 112)
#define WS_SSQ  (BC * KP)      // ws[7168] = sum of squares
#define P1_GRID 128
#define P1_NCHUNK (N_TOT / P1_GRID)   // 512
#define P1_TN   32
#define P1_ITERS (P1_NCHUNK / P1_TN)  // 16
#define PS1     36             // phase-1 LDS row stride (==4 mod 32: conflict-free, 16B-aligned)
#define P2_NCHUNK 128
#define P2_GRID (N_TOT / P2_NCHUNK)   // 512
#define PS2     116            // phase-2 LDS row stride (==4 mod 16)

// ---- gfx1250 helpers: async global->LDS DMA + native float atomics --------
__device__ __forceinline__ unsigned lds_off(const void* p) {
    return (unsigned)(unsigned long long)p;   // generic shared addr: low 32 bits = LDS byte offset
}
__device__ __forceinline__ void async_b128(unsigned lds, const void* g) {
    asm volatile("global_load_async_to_lds_b128 %0, %1, off" :: "v"(lds), "v"(g) : "memory");
}
__device__ __forceinline__ void async_b32(unsigned lds, const void* g) {
    asm volatile("global_load_async_to_lds_b32 %0, %1, off" :: "v"(lds), "v"(g) : "memory");
}
__device__ __forceinline__ void wait_async0() {
    asm volatile("s_wait_asynccnt 0x0" ::: "memory");
}
__device__ __forceinline__ void gatomic_fadd(float* p, float v) {
    asm volatile("global_atomic_add_f32 %0, %1, off scope:SCOPE_DEV" :: "v"(p), "v"(v) : "memory");
}

// ---------------------------------------------------------------- init ----
__global__ void spacy_init(float* ws) {
    int t = blockIdx.x * blockDim.x + threadIdx.x;
    for (int i = t; i <= WS_SSQ; i += blockDim.x * gridDim.x) ws[i] = 0.0f;
}

// ------------------------------------------------------------- phase 1 ----
// X[bc,k] = sum_n (y_pred - y)[bc,n] * mask[bc,n] * A_pinv[k,n]
__global__ void __launch_bounds__(256)
spacy_phase1(const float* __restrict__ yp, const float* __restrict__ yt,
             const int* __restrict__ mask, const float* __restrict__ pinv,
             float* __restrict__ ws) {
    __shared__ float rawYP[2][BC * P1_TN];   // 8KB x2
    __shared__ float rawYT[2][BC * P1_TN];   // 8KB x2
    __shared__ int   rawM [2][BC * P1_TN];   // 8KB x2
    __shared__ float Pkl  [2][KP * PS1];     // A_pinv slab [k][n], 16.1KB x2
    __shared__ float Dls  [BC * PS1];        // D slab [bc][n]
    __shared__ float Xls  [BC * KP];         // per-WG partial X

    const int t    = threadIdx.x;
    const int wave = t >> 5, lane = t & 31;
    const int mtile = wave & 3;              // 16 bc rows
    const int nhalf = wave >> 2;             // 16-n half of the 32-n slab
    const int l15 = lane & 15, hi = lane >> 4;
    const int nbase0 = blockIdx.x * P1_NCHUNK;

    for (int i = t; i < BC * KP; i += 256) Xls[i] = 0.0f;
    // zero the k=110,111 pad rows of both Pk buffers (async never writes them)
    if (t < 128) {
        const int b = t >> 6, rem = t & 63;
        Pkl[b][(K_REAL + (rem >> 5)) * PS1 + (rem & 31)] = 0.0f;
    }

    // issue one async batch: raw y/yhat/mask slabs (B128) + A_pinv slab (B128)
    auto issue_batch = [&](int buf, int nb) {
        for (int q = t; q < (BC * P1_TN) / 4; q += 256) {      // 512 quads
            const int r = q >> 3, c4 = (q & 7) * 4;
            const unsigned lo = (unsigned)(r * P1_TN + c4) * 4u;
            const int gi = r * N_TOT + nb + c4;
            async_b128(lds_off(&rawYP[buf][0]) + lo, yp   + gi);
            async_b128(lds_off(&rawYT[buf][0]) + lo, yt   + gi);
            async_b128(lds_off(&rawM [buf][0]) + lo, mask + gi);
        }
        for (int q = t; q < (K_REAL * P1_TN) / 4; q += 256) {  // 880 quads
            const int k = q >> 3, c4 = (q & 7) * 4;
            async_b128(lds_off(&Pkl[buf][0]) + (unsigned)(k * PS1 + c4) * 4u,
                       pinv + k * N_TOT + nb + c4);
        }
    };

    v8f acc[7];
#pragma unroll
    for (int kt = 0; kt < 7; ++kt) acc[kt] = (v8f){0,0,0,0,0,0,0,0};

    issue_batch(0, nbase0);

    for (int it = 0; it < P1_ITERS; ++it) {
        const int p = it & 1;
        wait_async0();           // own batch(p) complete
        __syncthreads();         // everyone's batch(p) visible; iter it-1 fully retired
        if (it + 1 < P1_ITERS) issue_batch(p ^ 1, nbase0 + (it + 1) * P1_TN);

        // D = (y_pred - y) * mask, LDS->LDS with padded stride
        for (int i = t; i < BC * P1_TN; i += 256) {
            const float d = (rawYP[p][i] - rawYT[p][i]) * (float)rawM[p][i];
            Dls[(i >> 5) * PS1 + (i & 31)] = d;
        }
        __syncthreads();

        for (int nn = nhalf * 16; nn < nhalf * 16 + 16; nn += 4) {
            const float* dp = &Dls[(mtile * 16 + l15) * PS1 + nn + 2 * hi];
            v2f a; a.x = dp[0]; a.y = dp[1];
#pragma unroll
            for (int kt = 0; kt < 7; ++kt) {
                const float* pp = &Pkl[p][(kt * 16 + l15) * PS1 + nn + 2 * hi];
                v2f b; b.x = pp[0]; b.y = pp[1];
                acc[kt] = __builtin_amdgcn_wmma_f32_16x16x4_f32(
                    false, a, false, b, (short)0, acc[kt], false, false);
            }
        }
    }
    __syncthreads();
    // combine the two n-half wave sets through LDS float atomics
#pragma unroll
    for (int kt = 0; kt < 7; ++kt)
#pragma unroll
        for (int j = 0; j < 8; ++j) {
            const int row = mtile * 16 + j + hi * 8;
            const int col = kt * 16 + l15;
            atomicAdd(&Xls[row * KP + col], acc[kt][j]);
        }
    __syncthreads();
    // one coalesced set of native global float atomics per workgroup
    for (int i = t; i < BC * KP; i += 256) gatomic_fadd(&ws[WS_X + i], Xls[i]);
}

// ------------------------------------------------------------- phase 2 ----
// ssq += sum_{bc,n} ( mask[bc,n] * sum_k A[n,k] * X[bc,k] )^2
__global__ void __launch_bounds__(256)
spacy_phase2(const int* __restrict__ mask, const float* __restrict__ Amat,
             float* __restrict__ ws) {
    __shared__ float Anl[P2_NCHUNK * PS2];   // A slab [n][k], 59.4KB
    __shared__ float Xls[BC * PS2];          // X [bc][k], 29.7KB
    __shared__ float Mkl[BC * P2_NCHUNK];    // mask slab [bc][n], 32.8KB
    __shared__ float red[8];

    const int t  = threadIdx.x;
    const int nb = blockIdx.x * P2_NCHUNK;

    // async-stage everything (per-lane LDS scatter handles the padded layouts)
    for (int i = t; i < BC * KP; i += 256) {                 // X, b32
        const int r = i / KP, c = i - r * KP;
        async_b32(lds_off(Xls) + (unsigned)(r * PS2 + c) * 4u, ws + WS_X + i);
    }
    for (int i = t; i < P2_NCHUNK * K_REAL; i += 256) {      // A, b32 (A is contiguous)
        const int n = i / K_REAL, k = i - n * K_REAL;
        async_b32(lds_off(Anl) + (unsigned)(n * PS2 + k) * 4u, Amat + nb * K_REAL + i);
    }
    for (int q = t; q < (BC * P2_NCHUNK) / 4; q += 256) {    // mask, b128
        const int r = q >> 5, c4 = (q & 31) * 4;
        async_b128(lds_off(Mkl) + (unsigned)(r * P2_NCHUNK + c4) * 4u,
                   mask + r * N_TOT + nb + c4);
    }
    if (t < 256) {                                           // zero A k-pads 110,111
        const int n = t >> 1;
        Anl[n * PS2 + K_REAL + (t & 1)] = 0.0f;
        Anl[(n + 128) * PS2 + K_REAL + (t & 1)] = 0.0f;      // rows n+128 in [128,256): past the 128-row slab
    }
    wait_async0();
    __syncthreads();

    const int wave = t >> 5, lane = t & 31;
    const int mtile = wave & 3;
    const int nsub  = wave >> 2;
    const int l15 = lane & 15, hi = lane >> 4;

    float ssq = 0.0f;
    for (int ntile = 0; ntile < 4; ++ntile) {
        const int ncol0 = nsub * 64 + ntile * 16;
        v8f acc = (v8f){0,0,0,0,0,0,0,0};
#pragma unroll 4
        for (int k0 = 0; k0 < KP; k0 += 4) {
            const float* xp2 = &Xls[(mtile * 16 + l15) * PS2 + k0 + 2 * hi];
            v2f a; a.x = xp2[0]; a.y = xp2[1];
            const float* bp = &Anl[(ncol0 + l15) * PS2 + k0 + 2 * hi];
            v2f b; b.x = bp[0]; b.y = bp[1];
            acc = __builtin_amdgcn_wmma_f32_16x16x4_f32(
                false, a, false, b, (short)0, acc, false, false);
        }
#pragma unroll
        for (int j = 0; j < 8; ++j) {
            const int row = mtile * 16 + j + hi * 8;
            const float v = acc[j] * Mkl[row * P2_NCHUNK + ncol0 + l15];
            ssq += v * v;
        }
    }
#pragma unroll
    for (int off = 16; off > 0; off >>= 1) ssq += __shfl_down(ssq, off, 32);
    if (lane == 0) red[wave] = ssq;
    __syncthreads();
    if (t == 0) {
        float s = 0.0f;
#pragma unroll
        for (int w = 0; w < 8; ++w) s += red[w];
        gatomic_fadd(&ws[WS_SSQ], s);
    }
}

// ------------------------------------------------------------ finalize ----
__global__ void spacy_final(const float* __restrict__ ws, float* __restrict__ out) {
    out[0] = ws[WS_SSQ] * (1.0f / (float)(BC * N_TOT));
}

extern "C" void kernel_launch(void* const* d_in, const int* in_sizes, int n_in,
                              void* d_out, int out_size, void* d_ws, size_t ws_size,
                              hipStream_t stream) {
    const float* y_pred = (const float*)d_in[0];
    const float* y      = (const float*)d_in[1];
    const int*   mask   = (const int*)d_in[2];
    const float* Amat   = (const float*)d_in[3];
    const float* A_pinv = (const float*)d_in[4];
    float* ws  = (float*)d_ws;
    float* out = (float*)d_out;

    spacy_init<<<1, 256, 0, stream>>>(ws);
    spacy_phase1<<<P1_GRID, 256, 0, stream>>>(y_pred, y, mask, A_pinv, ws);
    spacy_phase2<<<P2_GRID, 256, 0, stream>>>(mask, Amat, ws);
    spacy_final<<<1, 1, 0, stream>>>(ws, out);
}